// neigh_cognition_49890340110411
// MI455X (gfx1250) — compile-verified
//
#include <hip/hip_runtime.h>
#include <math.h>

#define IN_DIM 512
#define HID 1024
#define OUT 128
#define NN 32
#define ATT_DIM ((NN + 1) * HID)   // 33792
#define MID_DIM (HID * OUT)        // 131072

#define H_SPLIT 32
#define H_KS (ATT_DIM / H_SPLIT)   // 1056 (divisible by 4)
#define X_SPLIT 256
#define X_KS (MID_DIM / X_SPLIT)   // 512

typedef __attribute__((ext_vector_type(2))) float v2f;
typedef __attribute__((ext_vector_type(8))) float v8f;

// ---------------------------------------------------------------------------
// Core tile: D(16x16) = sum_k A(16x4) x B(4x16) using V_WMMA_F32_16X16X4_F32.
// A  : row-major M x K tile, row stride lda (weights, streamed nontemporal).
// Bt : row-major N x K tile, row stride ldbt. ldbt == 0 broadcasts a single
//      K-vector across all 16 N columns (GEMV mode).
// f32 A layout (ISA 7.12.2): lanes 0-15 hold K={0,1} in v0/v1 for M=lane;
// lanes 16-31 hold K={2,3}. B mirrors with N=lane&15. So each lane issues one
// b64 load per operand per K-step of 4.
// ---------------------------------------------------------------------------
__device__ __forceinline__ v8f wmma_tile_f32(const float* __restrict__ A, long lda,
                                             const float* __restrict__ Bt, long ldbt,
                                             int K) {
  const int lane = (int)(threadIdx.x & 31u);
  const int g    = (lane >> 4) * 2;     // K sub-offset within the 4-chunk
  const int idx  = lane & 15;           // M row for A / N column for Bt
  const float* ap = A  + (long)idx * lda  + g;
  const float* bp = Bt + (long)idx * ldbt + g;
  v8f acc = {};
  for (int k0 = 0; k0 < K; k0 += 4) {
    v2f a = __builtin_nontemporal_load((const v2f*)(ap + k0));
    v2f b = *(const v2f*)(bp + k0);
    acc = __builtin_amdgcn_wmma_f32_16x16x4_f32(false, a, false, b, (short)0,
                                                acc, false, false);
  }
  return acc;
}

// GEMV-mode store: every N column of D holds the same value; VGPR j carries
// rows m0+j (lanes 0-15) and m0+8+j (lanes 16-31). Lanes 0 and 16 write.
__device__ __forceinline__ void store_col(const v8f acc, float* __restrict__ y, int m0) {
  const int lane = (int)(threadIdx.x & 31u);
  if ((lane & 15) == 0) {
    const int base = m0 + (lane >> 4) * 8;
#pragma unroll
    for (int j = 0; j < 8; ++j) y[base + j] = acc[j];
  }
}

__device__ __forceinline__ void store_col_bias(const v8f acc, float* __restrict__ y,
                                               int m0, const float* __restrict__ bias) {
  const int lane = (int)(threadIdx.x & 31u);
  if ((lane & 15) == 0) {
    const int base = m0 + (lane >> 4) * 8;
#pragma unroll
    for (int j = 0; j < 8; ++j) y[base + j] = acc[j] + bias[base + j];
  }
}

// GEMM-mode store for K/V: out[(n0+n)*ldo + m], n = lane&15, m = m0+j(+8).
__device__ __forceinline__ void store_tile(const v8f acc, float* __restrict__ out, long ldo,
                                           int m0, int n0, const float* __restrict__ bias) {
  const int lane = (int)(threadIdx.x & 31u);
  const int n  = n0 + (lane & 15);
  const int mb = m0 + (lane >> 4) * 8;
  float* p = out + (long)n * ldo + mb;
#pragma unroll
  for (int j = 0; j < 8; ++j) p[j] = acc[j] + bias[mb + j];
}

// ---------------------------------------------------------------------------
// Kernel 1: q = Wq x + bq ; K = neigh Wk^T + bk ; V = neigh Wv^T + bv
// grid.x = 64 (q tiles) + 128 (K tiles) + 128 (V tiles) = 320, one wave each.
// ---------------------------------------------------------------------------
__global__ __launch_bounds__(32)
void k_qkv(const float* __restrict__ x, const float* __restrict__ neigh,
           const float* __restrict__ Wq, const float* __restrict__ bq,
           const float* __restrict__ Wk, const float* __restrict__ bk,
           const float* __restrict__ Wv, const float* __restrict__ bv,
           float* __restrict__ qws, float* __restrict__ Kws, float* __restrict__ Vws) {
  const int bid = (int)blockIdx.x;
  if (bid < HID / 16) {
    const int m0 = bid * 16;
    v8f acc = wmma_tile_f32(Wq + (long)m0 * IN_DIM, IN_DIM, x, 0, IN_DIM);
    store_col_bias(acc, qws, m0, bq);
  } else if (bid < HID / 16 + (HID / 16) * (NN / 16)) {
    const int t = bid - HID / 16;
    const int m0 = (t >> 1) * 16, i0 = (t & 1) * 16;
    v8f acc = wmma_tile_f32(Wk + (long)m0 * IN_DIM, IN_DIM,
                            neigh + (long)i0 * IN_DIM, IN_DIM, IN_DIM);
    store_tile(acc, Kws, HID, m0, i0, bk);
  } else {
    const int t = bid - HID / 16 - (HID / 16) * (NN / 16);
    const int m0 = (t >> 1) * 16, i0 = (t & 1) * 16;
    v8f acc = wmma_tile_f32(Wv + (long)m0 * IN_DIM, IN_DIM,
                            neigh + (long)i0 * IN_DIM, IN_DIM, IN_DIM);
    store_tile(acc, Vws, HID, m0, i0, bv);
  }
}

// ---------------------------------------------------------------------------
// Kernel 2: logits = (K q)/sqrt(NN); attend = softmax; build att (33792).
// One block, 1024 threads = 32 waves; wave i computes logit i.
// ---------------------------------------------------------------------------
__global__ __launch_bounds__(1024)
void k_attn(const float* __restrict__ qws, const float* __restrict__ Kws,
            const float* __restrict__ Vws, float* __restrict__ att) {
  __shared__ float s_logit[NN];
  __shared__ float s_attend[NN];
  const int tid = (int)threadIdx.x;
  const int lane = tid & 31, w = tid >> 5;

  float p = 0.f;
  for (int h = lane; h < HID; h += 32) p += Kws[w * HID + h] * qws[h];
#pragma unroll
  for (int d = 16; d > 0; d >>= 1) p += __shfl_xor(p, d, 32);
  if (lane == 0) s_logit[w] = p * 0.1767766952966369f;  // 1/sqrt(32)
  __syncthreads();

  if (w == 0) {
    float v = s_logit[lane];
    float mx = v;
#pragma unroll
    for (int d = 16; d > 0; d >>= 1) mx = fmaxf(mx, __shfl_xor(mx, d, 32));
    float e = __expf(v - mx);
    float s = e;
#pragma unroll
    for (int d = 16; d > 0; d >>= 1) s += __shfl_xor(s, d, 32);
    s_attend[lane] = e / s;
  }
  __syncthreads();

  att[tid] = qws[tid];
  for (int i = 0; i < NN; ++i)
    att[(1 + i) * HID + tid] = Vws[i * HID + tid] * s_attend[i];
}

// ---------------------------------------------------------------------------
// Kernel 3: split-K partials of Wo @ att.  grid = 64 mtiles x 32 ksplits.
// ---------------------------------------------------------------------------
__global__ __launch_bounds__(32)
void k_H_part(const float* __restrict__ Wo, const float* __restrict__ att,
              float* __restrict__ Hpart) {
  const int ks = (int)blockIdx.x & (H_SPLIT - 1);
  const int m0 = ((int)blockIdx.x >> 5) * 16;
  const long koff = (long)ks * H_KS;
  v8f acc = wmma_tile_f32(Wo + (long)m0 * ATT_DIM + koff, ATT_DIM, att + koff, 0, H_KS);
  store_col(acc, Hpart + ks * HID, m0);
}

__global__ __launch_bounds__(HID)
void k_H_reduce(const float* __restrict__ Hpart, const float* __restrict__ bo,
                float* __restrict__ H) {
  const int t = (int)threadIdx.x;
  float s = bo[t];
  for (int ks = 0; ks < H_SPLIT; ++ks) s += Hpart[ks * HID + t];
  H[t] = fmaxf(s, 0.f);
}

// ---------------------------------------------------------------------------
// Kernel 4: ihpre = W_ih H + b_ih + b_hh ; fcpre = Wfc H + bfc. (W_hh h0 == 0)
// ---------------------------------------------------------------------------
__global__ __launch_bounds__(32)
void k_C_gemv(const float* __restrict__ W_ih, const float* __restrict__ b_ih,
              const float* __restrict__ b_hh,
              const float* __restrict__ Wfc, const float* __restrict__ bfc,
              const float* __restrict__ H,
              float* __restrict__ ihpre, float* __restrict__ fcpre) {
  const int bid = (int)blockIdx.x;
  if (bid < HID / 16) {
    const int m0 = bid * 16;
    v8f acc = wmma_tile_f32(W_ih + (long)m0 * HID, HID, H, 0, HID);
    const int lane = (int)(threadIdx.x & 31u);
    if ((lane & 15) == 0) {
      const int base = m0 + (lane >> 4) * 8;
#pragma unroll
      for (int j = 0; j < 8; ++j)
        ihpre[base + j] = acc[j] + b_ih[base + j] + b_hh[base + j];
    }
  } else {
    const int m0 = (bid - HID / 16) * 16;
    v8f acc = wmma_tile_f32(Wfc + (long)m0 * HID, HID, H, 0, HID);
    store_col_bias(acc, fcpre, m0, bfc);
  }
}

__global__ __launch_bounds__(HID)
void k_C_elem(const float* __restrict__ ihpre, const float* __restrict__ fcpre,
              const float* __restrict__ noise,
              float* __restrict__ C_out, float* __restrict__ Cws) {
  const int t = (int)threadIdx.x;
  const float c = tanhf(ihpre[t]) + fcpre[t] * noise[t];
  C_out[t] = c;
  Cws[t]   = c;
}

// ---------------------------------------------------------------------------
// Kernel 5: mid = W1 C + b1 (dominant, 537 MB). 2048 blocks x 4 waves.
// ---------------------------------------------------------------------------
__global__ __launch_bounds__(128)
void k_mid(const float* __restrict__ W1, const float* __restrict__ b1,
           const float* __restrict__ Cws, float* __restrict__ mid) {
  const int mtile = (int)blockIdx.x * 4 + ((int)threadIdx.x >> 5);
  const int m0 = mtile * 16;
  v8f acc = wmma_tile_f32(W1 + (long)m0 * HID, HID, Cws, 0, HID);
  store_col_bias(acc, mid, m0, b1);
}

// ---------------------------------------------------------------------------
// Kernel 6: split-K partials of W2 @ mid. grid = 8 mtiles x 256 ksplits.
// ---------------------------------------------------------------------------
__global__ __launch_bounds__(32)
void k_xhat_part(const float* __restrict__ W2, const float* __restrict__ mid,
                 float* __restrict__ Xpart) {
  const int ks = (int)blockIdx.x & (X_SPLIT - 1);
  const int m0 = ((int)blockIdx.x >> 8) * 16;
  const long koff = (long)ks * X_KS;
  v8f acc = wmma_tile_f32(W2 + (long)m0 * MID_DIM + koff, MID_DIM, mid + koff, 0, X_KS);
  store_col(acc, Xpart + ks * OUT, m0);
}

__global__ __launch_bounds__(OUT)
void k_xhat_reduce(const float* __restrict__ Xpart, const float* __restrict__ b2,
                   float* __restrict__ xhat) {
  const int t = (int)threadIdx.x;
  float s = b2[t];
  for (int ks = 0; ks < X_SPLIT; ++ks) s += Xpart[ks * OUT + t];
  xhat[t] = s;
}

// ---------------------------------------------------------------------------
extern "C" void kernel_launch(void* const* d_in, const int* in_sizes, int n_in,
                              void* d_out, int out_size, void* d_ws, size_t ws_size,
                              hipStream_t stream) {
  const float* x     = (const float*)d_in[0];
  const float* neigh = (const float*)d_in[1];
  const float* Wq    = (const float*)d_in[2];
  const float* bq    = (const float*)d_in[3];
  const float* Wk    = (const float*)d_in[4];
  const float* bk    = (const float*)d_in[5];
  const float* Wv    = (const float*)d_in[6];
  const float* bv    = (const float*)d_in[7];
  const float* Wo    = (const float*)d_in[8];
  const float* bo    = (const float*)d_in[9];
  const float* W_ih  = (const float*)d_in[10];
  const float* b_ih  = (const float*)d_in[11];
  // d_in[12] = W_hh (unused: h0 == 0), d_in[13] = b_hh (used)
  const float* b_hh  = (const float*)d_in[13];
  const float* Wfc   = (const float*)d_in[14];
  const float* bfc   = (const float*)d_in[15];
  const float* W1    = (const float*)d_in[16];
  const float* b1    = (const float*)d_in[17];
  const float* W2    = (const float*)d_in[18];
  const float* b2    = (const float*)d_in[19];
  const float* noise = (const float*)d_in[20];
  float* out = (float*)d_out;

  // workspace layout (floats)
  float* ws    = (float*)d_ws;
  float* qws   = ws;                     // 1024
  float* Kws   = qws   + HID;            // 32768
  float* Vws   = Kws   + NN * HID;       // 32768
  float* att   = Vws   + NN * HID;       // 33792
  float* Hpart = att   + ATT_DIM;        // 32768
  float* Hws   = Hpart + H_SPLIT * HID;  // 1024
  float* ihpre = Hws   + HID;            // 1024
  float* fcpre = ihpre + HID;            // 1024
  float* Cws   = fcpre + HID;            // 1024
  float* midws = Cws   + HID;            // 131072
  float* Xpart = midws + MID_DIM;        // 32768
  (void)in_sizes; (void)n_in; (void)out_size; (void)ws_size;

  k_qkv<<<HID / 16 + 2 * (HID / 16) * (NN / 16), 32, 0, stream>>>(
      x, neigh, Wq, bq, Wk, bk, Wv, bv, qws, Kws, Vws);
  k_attn<<<1, 1024, 0, stream>>>(qws, Kws, Vws, att);
  k_H_part<<<(HID / 16) * H_SPLIT, 32, 0, stream>>>(Wo, att, Hpart);
  k_H_reduce<<<1, HID, 0, stream>>>(Hpart, bo, Hws);
  k_C_gemv<<<2 * (HID / 16), 32, 0, stream>>>(W_ih, b_ih, b_hh, Wfc, bfc, Hws,
                                              ihpre, fcpre);
  k_C_elem<<<1, HID, 0, stream>>>(ihpre, fcpre, noise, out, Cws);
  k_mid<<<(MID_DIM / 16) / 4, 128, 0, stream>>>(W1, b1, Cws, midws);
  k_xhat_part<<<(OUT / 16) * X_SPLIT, 32, 0, stream>>>(W2, midws, Xpart);
  k_xhat_reduce<<<1, OUT, 0, stream>>>(Xpart, b2, out + HID);
}